// Mask_c_35905926595057
// MI455X (gfx1250) — compile-verified
//
#include <hip/hip_runtime.h>
#include <math.h>

typedef __attribute__((ext_vector_type(2))) float v2f;
typedef __attribute__((ext_vector_type(8))) float v8f;

#define GATE_EPS_F 0.66667f
#define SCALE_QK 0.044194173824159216f   // 512^-0.5

// ---------------------------------------------------------------------------
// WMMA helpers: fp32 16x16x4, D = A*B + C.
// A (16x4, MxK): lanes 0-15 -> M=lane, VGPR0/1 = K {k,k+1}; lanes 16-31 -> K {k+2,k+3}
// B (4x16, KxN): lanes 0-15 -> N=lane, VGPR0/1 = K {k,k+1}; lanes 16-31 -> K {k+2,k+3}
// C/D (16x16):   VGPR g, lane l -> row = g + 8*(l>>4), col = l&15
// ---------------------------------------------------------------------------
__device__ __forceinline__ v8f wmma_f32(v2f a, v2f b, v8f c) {
  return __builtin_amdgcn_wmma_f32_16x16x4_f32(false, a, false, b, (short)0, c,
                                               false, false);
}

// 16x16 tile of A(MxK, lda=K) @ W(NxK, ldw=K)^T   (i.e. A @ W.T)
__device__ __forceinline__ v8f tile_bt(const float* __restrict__ A,
                                       const float* __restrict__ W,
                                       int K, int m0, int n0, int lane) {
  const int half = lane >> 4;
  const int lm   = lane & 15;
  v8f acc = {0.f, 0.f, 0.f, 0.f, 0.f, 0.f, 0.f, 0.f};
  const float* ar = A + (size_t)(m0 + lm) * K + 2 * half;
  const float* wr = W + (size_t)(n0 + lm) * K + 2 * half;
  for (int k = 0; k < K; k += 4) {
    v2f a = *(const v2f*)(ar + k);
    v2f b = *(const v2f*)(wr + k);
    acc = wmma_f32(a, b, acc);
  }
  return acc;
}

// 16x16 tile of A(MxK, lda=K) @ B(KxN, ldb=N)     (B row-major, not transposed)
__device__ __forceinline__ v8f tile_bn(const float* __restrict__ A,
                                       const float* __restrict__ B,
                                       int K, int ldb, int m0, int n0, int lane) {
  const int half = lane >> 4;
  const int lm   = lane & 15;
  v8f acc = {0.f, 0.f, 0.f, 0.f, 0.f, 0.f, 0.f, 0.f};
  const float* ar = A + (size_t)(m0 + lm) * K + 2 * half;
  for (int k = 0; k < K; k += 4) {
    v2f a = *(const v2f*)(ar + k);
    v2f b;
    b.x = B[(size_t)(k + 2 * half)     * ldb + n0 + lm];
    b.y = B[(size_t)(k + 2 * half + 1) * ldb + n0 + lm];
    acc = wmma_f32(a, b, acc);
  }
  return acc;
}

__device__ __forceinline__ float sigmoidf_(float x) {
  return 1.f / (1.f + expf(-x));
}

// ---------------------------------------------------------------------------
// ctx = mean(x, axis=(2,3)) : one wave per (b,c) row of 196 contiguous floats
// ---------------------------------------------------------------------------
__global__ void k_reduce_x(const float* __restrict__ x, float* __restrict__ ctx) {
  const int wave = (blockIdx.x * blockDim.x + threadIdx.x) >> 5;
  const int lane = threadIdx.x & 31;
  const float* p = x + (size_t)wave * 196;
  float s = 0.f;
  for (int i = lane; i < 196; i += 32) s += p[i];
  for (int off = 16; off > 0; off >>= 1) s += __shfl_xor(s, off, 32);
  if (lane == 0) ctx[wave] = s * (1.0f / 196.0f);
}

// ---------------------------------------------------------------------------
// Row LayerNorm over D=512, one block per row (64 rows)
// ---------------------------------------------------------------------------
__global__ void k_ln(const float* __restrict__ X, const float* __restrict__ g,
                     const float* __restrict__ b, float* __restrict__ Y) {
  const int D = 512;
  const int r = blockIdx.x;
  const float* x = X + (size_t)r * D;
  float s = 0.f, s2 = 0.f;
  for (int i = threadIdx.x; i < D; i += blockDim.x) {
    float v = x[i]; s += v; s2 += v * v;
  }
  for (int off = 16; off > 0; off >>= 1) {
    s  += __shfl_xor(s,  off, 32);
    s2 += __shfl_xor(s2, off, 32);
  }
  __shared__ float r1[8], r2[8];
  const int wave = threadIdx.x >> 5, lane = threadIdx.x & 31;
  if (lane == 0) { r1[wave] = s; r2[wave] = s2; }
  __syncthreads();
  if (threadIdx.x == 0) {
    float a = 0.f, a2 = 0.f;
    for (int w = 0; w < 8; ++w) { a += r1[w]; a2 += r2[w]; }
    r1[0] = a; r2[0] = a2;
  }
  __syncthreads();
  const float mean = r1[0] / D;
  const float var  = r2[0] / D - mean * mean;
  const float inv  = rsqrtf(var + 1e-5f);
  for (int i = threadIdx.x; i < D; i += blockDim.x)
    Y[(size_t)r * D + i] = (x[i] - mean) * inv * g[i] + b[i];
}

// ---------------------------------------------------------------------------
// Generic C(64xN) = A(64xK) @ W(NxK)^T + bias, epilogue: 0 none, 1 relu,
// 2 add extra. tiles = 4*(N/16), 8 waves/block.
// ---------------------------------------------------------------------------
template <int EPI>
__global__ void k_gemm_bt(const float* __restrict__ A, const float* __restrict__ W,
                          const float* __restrict__ bias,
                          const float* __restrict__ extra,
                          float* __restrict__ C, int N, int K) {
  const int wave = threadIdx.x >> 5, lane = threadIdx.x & 31;
  const int tile = blockIdx.x * (blockDim.x >> 5) + wave;
  const int ntn  = N >> 4;
  const int mt = tile / ntn, nt = tile % ntn;
  v8f acc = tile_bt(A, W, K, mt * 16, nt * 16, lane);
  const int half = lane >> 4, lm = lane & 15;
  const int col = nt * 16 + lm;
  const float bv = bias ? bias[col] : 0.f;
  for (int g = 0; g < 8; ++g) {
    const int row = mt * 16 + g + 8 * half;
    float v = acc[g] + bv;
    if (EPI == 1) v = fmaxf(v, 0.f);
    if (EPI == 2) v += extra[(size_t)row * N + col];
    C[(size_t)row * N + col] = v;
  }
}

// C(64xN) = A(64xK) @ B(KxN), no bias (updates = attn @ v)
__global__ void k_gemm_bn(const float* __restrict__ A, const float* __restrict__ B,
                          float* __restrict__ C, int N, int K) {
  const int wave = threadIdx.x >> 5, lane = threadIdx.x & 31;
  const int tile = blockIdx.x * (blockDim.x >> 5) + wave;
  const int ntn  = N >> 4;
  const int mt = tile / ntn, nt = tile % ntn;
  v8f acc = tile_bn(A, B, K, N, mt * 16, nt * 16, lane);
  const int half = lane >> 4, lm = lane & 15;
  for (int g = 0; g < 8; ++g)
    C[(size_t)(mt * 16 + g + 8 * half) * N + nt * 16 + lm] = acc[g];
}

// ---------------------------------------------------------------------------
// attn = softmax(q @ k^T * SCALE, axis=1) : 64x64, one block
// ---------------------------------------------------------------------------
__global__ void k_attn(const float* __restrict__ q, const float* __restrict__ kk,
                       float* __restrict__ attn) {
  __shared__ float sc[64 * 64];
  const int wave = threadIdx.x >> 5, lane = threadIdx.x & 31;
  const int half = lane >> 4, lm = lane & 15;
  for (int t = wave; t < 16; t += 8) {
    const int mt = t >> 2, nt = t & 3;
    v8f acc = tile_bt(q, kk, 512, mt * 16, nt * 16, lane);
    for (int g = 0; g < 8; ++g)
      sc[(mt * 16 + g + 8 * half) * 64 + nt * 16 + lm] = acc[g] * SCALE_QK;
  }
  __syncthreads();
  if (threadIdx.x < 64) {
    float* row = sc + threadIdx.x * 64;
    float mx = -INFINITY;
    for (int j = 0; j < 64; ++j) mx = fmaxf(mx, row[j]);
    float s = 0.f;
    for (int j = 0; j < 64; ++j) { float e = expf(row[j] - mx); row[j] = e; s += e; }
    const float inv = 1.f / s;
    for (int j = 0; j < 64; ++j) attn[threadIdx.x * 64 + j] = row[j] * inv;
  }
}

// ---------------------------------------------------------------------------
// Fused GRU cell: h = (1-z)*n + z*h_prev, per 16x16 output tile, 6 accumulators
// wih/whh are (1536,512); gate g uses rows [g*512 .. g*512+511].
// ---------------------------------------------------------------------------
__global__ void k_gru(const float* __restrict__ upd, const float* __restrict__ hprev,
                      const float* __restrict__ wih, const float* __restrict__ whh,
                      const float* __restrict__ bih, const float* __restrict__ bhh,
                      float* __restrict__ h) {
  const int D = 512;
  const int wave = threadIdx.x >> 5, lane = threadIdx.x & 31;
  const int tile = blockIdx.x * (blockDim.x >> 5) + wave;   // 128 tiles
  const int mt = tile >> 5, nt = tile & 31;
  const int m0 = mt * 16, n0 = nt * 16;
  v8f ir = tile_bt(upd,   wih,                      D, m0, n0, lane);
  v8f iz = tile_bt(upd,   wih + (size_t)D * D,      D, m0, n0, lane);
  v8f in = tile_bt(upd,   wih + (size_t)2 * D * D,  D, m0, n0, lane);
  v8f hr = tile_bt(hprev, whh,                      D, m0, n0, lane);
  v8f hz = tile_bt(hprev, whh + (size_t)D * D,      D, m0, n0, lane);
  v8f hn = tile_bt(hprev, whh + (size_t)2 * D * D,  D, m0, n0, lane);
  const int half = lane >> 4, lm = lane & 15;
  const int col = n0 + lm;
  const float b_ir = bih[col], b_iz = bih[512 + col], b_in = bih[1024 + col];
  const float b_hr = bhh[col], b_hz = bhh[512 + col], b_hn = bhh[1024 + col];
  for (int g = 0; g < 8; ++g) {
    const int row = m0 + g + 8 * half;
    const float hp = hprev[(size_t)row * D + col];
    const float r  = sigmoidf_(ir[g] + b_ir + hr[g] + b_hr);
    const float z  = sigmoidf_(iz[g] + b_iz + hz[g] + b_hz);
    const float nn = tanhf(in[g] + b_in + r * (hn[g] + b_hn));
    h[(size_t)row * D + col] = (1.f - z) * nn + z * hp;
  }
}

// ---------------------------------------------------------------------------
// c = relu(BN_axis0(ctxfc + slots)); one thread per column (512 cols)
// ---------------------------------------------------------------------------
__global__ void k_bn(const float* __restrict__ ctxfc, const float* __restrict__ slots,
                     const float* __restrict__ g, const float* __restrict__ b,
                     float* __restrict__ c) {
  const int col = blockIdx.x * blockDim.x + threadIdx.x;
  if (col >= 512) return;
  float s = 0.f, s2 = 0.f;
  for (int r = 0; r < 64; ++r) {
    const float v = ctxfc[(size_t)r * 512 + col] + slots[(size_t)r * 512 + col];
    s += v; s2 += v * v;
  }
  const float m   = s / 64.f;
  const float var = s2 / 64.f - m * m;
  const float inv = rsqrtf(var + 1e-5f);
  for (int r = 0; r < 64; ++r) {
    const float v = ctxfc[(size_t)r * 512 + col] + slots[(size_t)r * 512 + col];
    c[(size_t)r * 512 + col] = fmaxf((v - m) * inv * g[col] + b[col], 0.f);
  }
}

__global__ void k_init_out(float* __restrict__ norm, float* __restrict__ norm_t) {
  if (threadIdx.x < 64) norm[threadIdx.x] = 0.f;
  if (threadIdx.x == 64) norm_t[0] = 2048.f;
}

// ---------------------------------------------------------------------------
// logits = c @ conv_w^T + logit(u); mask = hard gate; norm += row sums.
// M=64, N=2048, K=512 -> 512 tiles, 64 blocks x 8 waves.
// ---------------------------------------------------------------------------
__global__ void k_logits(const float* __restrict__ c, const float* __restrict__ convw,
                         const float* __restrict__ u, float* __restrict__ mask,
                         float* __restrict__ norm) {
  const int wave = threadIdx.x >> 5, lane = threadIdx.x & 31;
  const int tile = blockIdx.x * (blockDim.x >> 5) + wave;
  const int mt = tile >> 7, nt = tile & 127;
  v8f acc = tile_bt(c, convw, 512, mt * 16, nt * 16, lane);
  const int half = lane >> 4, lm = lane & 15;
  const int col = nt * 16 + lm;
  for (int g = 0; g < 8; ++g) {
    const int row = mt * 16 + g + 8 * half;
    const float uu = u[(size_t)row * 2048 + col];
    const float gl = logf(uu + 1e-8f) - logf(1.f - uu + 1e-8f);
    const float l  = acc[g] + gl;
    const float soft = sigmoidf_(l / GATE_EPS_F);
    const float hard = (soft >= 0.5f) ? 1.f : 0.f;
    mask[(size_t)row * 2048 + col] = hard;
    float p = hard;                           // reduce over the 16 lanes of this half
    for (int off = 1; off < 16; off <<= 1) p += __shfl_xor(p, off, 32);
    if (lm == 0) atomicAdd(&norm[row], p);
  }
}

__global__ void k_copy(const float* __restrict__ src, float* __restrict__ dst, int n) {
  const int i = blockIdx.x * blockDim.x + threadIdx.x;
  if (i < n) dst[i] = src[i];
}

// ---------------------------------------------------------------------------
extern "C" void kernel_launch(void* const* d_in, const int* in_sizes, int n_in,
                              void* d_out, int out_size, void* d_ws, size_t ws_size,
                              hipStream_t stream) {
  (void)in_sizes; (void)n_in; (void)out_size; (void)ws_size;
  const float* x       = (const float*)d_in[0];
  const float* txt     = (const float*)d_in[1];
  const float* u       = (const float*)d_in[2];
  const float* fc1_w   = (const float*)d_in[3];
  const float* fc1_b   = (const float*)d_in[4];
  const float* q_w     = (const float*)d_in[5];
  const float* q_b     = (const float*)d_in[6];
  const float* kw      = (const float*)d_in[7];
  const float* kbias   = (const float*)d_in[8];
  const float* vw      = (const float*)d_in[9];
  const float* vbias   = (const float*)d_in[10];
  const float* gru_wih = (const float*)d_in[11];
  const float* gru_whh = (const float*)d_in[12];
  const float* gru_bih = (const float*)d_in[13];
  const float* gru_bhh = (const float*)d_in[14];
  const float* mlp_w1  = (const float*)d_in[15];
  const float* mlp_b1  = (const float*)d_in[16];
  const float* mlp_w2  = (const float*)d_in[17];
  const float* mlp_b2  = (const float*)d_in[18];
  const float* ln_in_g = (const float*)d_in[19];
  const float* ln_in_b = (const float*)d_in[20];
  const float* ln_s_g  = (const float*)d_in[21];
  const float* ln_s_b  = (const float*)d_in[22];
  const float* ln_ff_g = (const float*)d_in[23];
  const float* ln_ff_b = (const float*)d_in[24];
  const float* bn_g    = (const float*)d_in[25];
  const float* bn_b    = (const float*)d_in[26];
  const float* conv_w  = (const float*)d_in[27];

  float* W     = (float*)d_ws;
  float* ctx   = W;               // 64*2048
  float* ctxfc = W + 131072;      // 64*512
  float* slots = W + 163840;
  float* pano  = W + 196608;
  float* kb    = W + 229376;
  float* vb    = W + 262144;
  float* sb    = W + 294912;
  float* qb    = W + 327680;
  float* att   = W + 360448;      // 64*64
  float* upd   = W + 364544;
  float* hb    = W + 397312;
  float* tb    = W + 430080;      // total ~1.77 MB

  float* mask   = (float*)d_out;
  float* norm   = mask + 64 * 2048;
  float* norm_t = norm + 64;

  // 1) ctx = mean over HxW (memory-bound: ~103 MB of x)
  k_reduce_x<<<16384, 256, 0, stream>>>(x, ctx);
  // 2) ctxfc = ctx @ fc1_w.T + fc1_b  (M64 N512 K2048 -> 128 tiles)
  k_gemm_bt<0><<<16, 256, 0, stream>>>(ctx, fc1_w, fc1_b, nullptr, ctxfc, 512, 2048);
  k_copy<<<128, 256, 0, stream>>>(ctxfc, slots, 32768);
  // 3) pano = LN(txt); k/v projections
  k_ln<<<64, 256, 0, stream>>>(txt, ln_in_g, ln_in_b, pano);
  k_gemm_bt<0><<<16, 256, 0, stream>>>(pano, kw, kbias, nullptr, kb, 512, 512);
  k_gemm_bt<0><<<16, 256, 0, stream>>>(pano, vw, vbias, nullptr, vb, 512, 512);
  // 4) slot-attention loop
  for (int it = 0; it < 3; ++it) {
    k_ln<<<64, 256, 0, stream>>>(slots, ln_s_g, ln_s_b, sb);
    k_gemm_bt<0><<<16, 256, 0, stream>>>(sb, q_w, q_b, nullptr, qb, 512, 512);
    k_attn<<<1, 256, 0, stream>>>(qb, kb, att);
    k_gemm_bn<<<16, 256, 0, stream>>>(att, vb, upd, 512, 64);
    k_gru<<<16, 256, 0, stream>>>(upd, slots, gru_wih, gru_whh, gru_bih, gru_bhh, hb);
    k_ln<<<64, 256, 0, stream>>>(hb, ln_ff_g, ln_ff_b, sb);
    k_gemm_bt<1><<<16, 256, 0, stream>>>(sb, mlp_w1, mlp_b1, nullptr, tb, 512, 512);
    k_gemm_bt<2><<<16, 256, 0, stream>>>(tb, mlp_w2, mlp_b2, hb, slots, 512, 512);
  }
  // 5) batch-norm + relu, then gated logits
  k_bn<<<2, 256, 0, stream>>>(ctxfc, slots, bn_g, bn_b, sb);
  k_init_out<<<1, 128, 0, stream>>>(norm, norm_t);
  k_logits<<<64, 256, 0, stream>>>(sb, conv_w, u, mask, norm);
}